// ChebNet_8993661518598
// MI455X (gfx1250) — compile-verified
//
#include <hip/hip_runtime.h>
#include <hip/hip_bf16.h>

typedef __attribute__((ext_vector_type(2))) float v2f;
typedef __attribute__((ext_vector_type(4))) float v4f;
typedef __attribute__((ext_vector_type(8))) float v8f;

#define NN 100000
#define NE 3200000
#define CIN 128

// ---------------- utility kernels ----------------

__global__ void zero_kernel(float* __restrict__ p, long long n) {
  long long i = ((long long)blockIdx.x * blockDim.x + threadIdx.x) * 4;
  if (i < n) {
    v4f z = {0.f, 0.f, 0.f, 0.f};
    *(v4f*)(p + i) = z;
  }
}

__global__ void negcopy_kernel(const float* __restrict__ s, float* __restrict__ d, long long n) {
  long long i = ((long long)blockIdx.x * blockDim.x + threadIdx.x) * 4;
  if (i < n) {
    v4f v = *(const v4f*)(s + i);
    *(v4f*)(d + i) = -v;
  }
}

// ---------------- graph normalization ----------------

__global__ void deg_kernel(const long long* __restrict__ row, float* __restrict__ deg) {
  int e = blockIdx.x * blockDim.x + threadIdx.x;
  if (e < NE) unsafeAtomicAdd(&deg[(int)row[e]], 1.0f);
}

__global__ void dis_kernel(float* __restrict__ deg) {
  int i = blockIdx.x * blockDim.x + threadIdx.x;
  if (i < NN) {
    float d = deg[i];
    deg[i] = d > 0.f ? rsqrtf(fmaxf(d, 1.f)) : 0.f;
  }
}

__global__ void ew_kernel(const long long* __restrict__ row, const long long* __restrict__ col,
                          const float* __restrict__ dis, float* __restrict__ w) {
  int e = blockIdx.x * blockDim.x + threadIdx.x;
  if (e < NE) w[e] = -dis[(int)row[e]] * dis[(int)col[e]];
}

// ---------------- SpMM: out[row] += scale * w_e * t[col]  (128 channels) ----------------
// One wave per edge, 4 channels per lane (float4 gather, fp32 HW atomics; operands
// are L2-resident: the 51.2 MB feature matrix fits the 192 MB global L2).

__global__ __launch_bounds__(256) void spmm_kernel(
    const float* __restrict__ t, float* __restrict__ out,
    const float* __restrict__ w, const long long* __restrict__ row,
    const long long* __restrict__ col, float scale) {
  long long gid = (long long)blockIdx.x * blockDim.x + threadIdx.x;
  long long e = gid >> 5;
  int lane = threadIdx.x & 31;
  if (e >= NE) return;
  int r = (int)row[e];
  int c = (int)col[e];
  float s = scale * w[e];
  v4f v = *(const v4f*)(t + (long long)c * CIN + lane * 4);
  float* dst = out + (long long)r * CIN + lane * 4;
  unsafeAtomicAdd(dst + 0, s * v.x);
  unsafeAtomicAdd(dst + 1, s * v.y);
  unsafeAtomicAdd(dst + 2, s * v.z);
  unsafeAtomicAdd(dst + 3, s * v.w);
}

// ---------------- fused ChebConv GEMM:  out = [T0|T1|T2|T3] (N x 512) @ W (512 x OUTC) + b ----
// fp32 WMMA 16x16x4. One wave per 16x16 output tile, K=512 in steps of 4 (128 WMMAs).
// A frag (16x4 f32): lanes 0-15 hold M=lane K={k,k+1}; lanes 16-31 hold K={k+2,k+3}.
// B frag (4x16 f32): lanes 0-15 hold N=lane K={k,k+1}; lanes 16-31 hold K={k+2,k+3}.
// C/D (16x16 f32): VGPR v, lane l -> M = v + (l>=16 ? 8 : 0), Ncol = l & 15.

template <int OUTC, bool RELU>
__global__ __launch_bounds__(256) void cheb_gemm_kernel(
    const float* __restrict__ T0, const float* __restrict__ T1,
    const float* __restrict__ T2, const float* __restrict__ T3,
    const float* __restrict__ W, const float* __restrict__ bias,
    float* __restrict__ out) {
  constexpr int TN = OUTC / 16;
  int wave = threadIdx.x >> 5;
  int tile = blockIdx.x * 8 + wave;
  if (tile >= (NN / 16) * TN) return;
  int tm = tile / TN;
  int tn = tile % TN;
  int lane = threadIdx.x & 31;
  int hl = lane >> 4;     // which K half-pair this lane provides
  int l16 = lane & 15;

  const float* As[4] = {T0, T1, T2, T3};
  const long long arow = (long long)(tm * 16 + l16) * CIN + hl * 2;
  const int bcol = tn * 16 + l16;

  v8f acc = {0.f, 0.f, 0.f, 0.f, 0.f, 0.f, 0.f, 0.f};

#pragma unroll
  for (int seg = 0; seg < 4; ++seg) {
    const float* __restrict__ A = As[seg] + arow;
    const float* __restrict__ B = W + (long long)(seg * CIN + hl * 2) * OUTC + bcol;
#pragma unroll
    for (int kk = 0; kk < CIN / 4; ++kk) {
      v2f a = {A[kk * 4 + 0], A[kk * 4 + 1]};
      v2f b = {B[(long long)(kk * 4 + 0) * OUTC], B[(long long)(kk * 4 + 1) * OUTC]};
      acc = __builtin_amdgcn_wmma_f32_16x16x4_f32(false, a, false, b, (short)0, acc,
                                                  false, false);
    }
  }

  float bv = bias[tn * 16 + l16];
  int row0 = tm * 16 + hl * 8;
  float* o = out + (long long)row0 * OUTC + tn * 16 + l16;
#pragma unroll
  for (int v = 0; v < 8; ++v) {
    float r = acc[v] + bv;
    if (RELU) r = fmaxf(r, 0.f);
    o[(long long)v * OUTC] = r;
  }
}

// ---------------- log_softmax over 64 channels, one wave per node, in place ----------------

__global__ __launch_bounds__(256) void logsoftmax_kernel(float* __restrict__ z) {
  long long gid = (long long)blockIdx.x * blockDim.x + threadIdx.x;
  long long r = gid >> 5;
  int lane = threadIdx.x & 31;
  if (r >= NN) return;
  float* p = z + r * 64 + lane * 2;
  float a = p[0], b = p[1];
  float m = fmaxf(a, b);
#pragma unroll
  for (int o = 16; o > 0; o >>= 1) m = fmaxf(m, __shfl_xor(m, o, 32));
  float s = expf(a - m) + expf(b - m);
#pragma unroll
  for (int o = 16; o > 0; o >>= 1) s += __shfl_xor(s, o, 32);
  float ls = m + logf(s);
  p[0] = a - ls;
  p[1] = b - ls;
}

// ---------------- launcher ----------------

extern "C" void kernel_launch(void* const* d_in, const int* in_sizes, int n_in,
                              void* d_out, int out_size, void* d_ws, size_t ws_size,
                              hipStream_t stream) {
  (void)in_sizes; (void)n_in; (void)out_size; (void)ws_size;

  const float* x  = (const float*)d_in[0];            // N x 128
  const float* W1 = (const float*)d_in[1];            // 4 x 128 x 128 == 512 x 128
  const float* b1 = (const float*)d_in[2];            // 128
  const float* W2 = (const float*)d_in[3];            // 4 x 128 x 64  == 512 x 64
  const float* b2 = (const float*)d_in[4];            // 64
  const long long* ei  = (const long long*)d_in[5];   // int64 [2, E]
  const long long* row = ei;
  const long long* col = ei + NE;

  float* ws  = (float*)d_ws;
  float* wE  = ws;                                // E edge weights        (12.8 MB)
  float* dis = wE + NE;                           // N  deg -> dis          (0.4 MB)
  float* T1  = dis + NN;                          // N x 128               (51.2 MB)
  float* T2  = T1 + (long long)NN * CIN;          // N x 128
  float* T3  = T2 + (long long)NN * CIN;          // N x 128
  float* h   = T3 + (long long)NN * CIN;          // N x 128 (layer-1 out)
  float* oz  = (float*)d_out;                     // N x 64

  const long long NF = (long long)NN * CIN;
  dim3 blk(256);
  auto gsz = [](long long nthreads) {
    return dim3((unsigned)((nthreads + 255) / 256));
  };

  // normalization weights
  zero_kernel<<<gsz(NN / 4), blk, 0, stream>>>(dis, NN);
  deg_kernel<<<gsz(NE), blk, 0, stream>>>(row, dis);
  dis_kernel<<<gsz(NN), blk, 0, stream>>>(dis);
  ew_kernel<<<gsz(NE), blk, 0, stream>>>(row, col, dis, wE);

  const long long spmm_threads = (long long)NE * 32;

  // ---- layer 1: T0 = x ----
  zero_kernel<<<gsz(NF / 4), blk, 0, stream>>>(T1, NF);
  spmm_kernel<<<gsz(spmm_threads), blk, 0, stream>>>(x, T1, wE, row, col, 1.0f);
  negcopy_kernel<<<gsz(NF / 4), blk, 0, stream>>>(x, T2, NF);       // T2 = -T0
  spmm_kernel<<<gsz(spmm_threads), blk, 0, stream>>>(T1, T2, wE, row, col, 2.0f);
  negcopy_kernel<<<gsz(NF / 4), blk, 0, stream>>>(T1, T3, NF);      // T3 = -T1
  spmm_kernel<<<gsz(spmm_threads), blk, 0, stream>>>(T2, T3, wE, row, col, 2.0f);
  // h = relu([x|T1|T2|T3] @ W1 + b1) : 50000 tiles / 8 waves
  cheb_gemm_kernel<128, true><<<dim3(6250), blk, 0, stream>>>(x, T1, T2, T3, W1, b1, h);

  // ---- layer 2: T0 = h ----
  zero_kernel<<<gsz(NF / 4), blk, 0, stream>>>(T1, NF);
  spmm_kernel<<<gsz(spmm_threads), blk, 0, stream>>>(h, T1, wE, row, col, 1.0f);
  negcopy_kernel<<<gsz(NF / 4), blk, 0, stream>>>(h, T2, NF);
  spmm_kernel<<<gsz(spmm_threads), blk, 0, stream>>>(T1, T2, wE, row, col, 2.0f);
  negcopy_kernel<<<gsz(NF / 4), blk, 0, stream>>>(T1, T3, NF);
  spmm_kernel<<<gsz(spmm_threads), blk, 0, stream>>>(T2, T3, wE, row, col, 2.0f);
  // z = [h|T1|T2|T3] @ W2 + b2 : 25000 tiles / 8 waves
  cheb_gemm_kernel<64, false><<<dim3(3125), blk, 0, stream>>>(h, T1, T2, T3, W2, b2, oz);

  // log_softmax in place on d_out
  logsoftmax_kernel<<<gsz((long long)NN * 32), blk, 0, stream>>>(oz);
}